// Gaussians_36335423324561
// MI455X (gfx1250) — compile-verified
//
#include <hip/hip_runtime.h>
#include <hip/hip_bf16.h>

#define TPB 256

// Pointee type the gfx1250 async-LDS builtins expect: int4 vector.
typedef int v4i __attribute__((vector_size(16)));
typedef __attribute__((address_space(1))) v4i gv4i;  // global (AS1)
typedef __attribute__((address_space(3))) v4i lv4i;  // LDS (AS3)

// gfx1250 async Global<->LDS DMA builtins, guarded so the file always compiles.
#if defined(__gfx1250__) &&                                                    \
    __has_builtin(__builtin_amdgcn_global_load_async_to_lds_b128) &&           \
    __has_builtin(__builtin_amdgcn_global_store_async_from_lds_b128) &&        \
    __has_builtin(__builtin_amdgcn_s_wait_asynccnt)
#define USE_ASYNC_LDS 1
#else
#define USE_ASYNC_LDS 0
#endif

__device__ __forceinline__ void cov_from_quat(float r, float x, float y, float z,
                                              float s0, float s1, float s2,
                                              float c[9]) {
    float nn = r * r + x * x + y * y + z * z;
    // inv = 1/max(||q||, 1e-12) == rsq(max(||q||^2, 1e-24)); single v_rsq_f32
    // instead of the IEEE sqrt+div expansion (~25 VALU ops). ~1 ulp accuracy,
    // far inside tolerance for degree-4 polynomial outputs.
#if defined(__HIP_DEVICE_COMPILE__) && __has_builtin(__builtin_amdgcn_rsqf)
    float inv = __builtin_amdgcn_rsqf(fmaxf(nn, 1e-24f));
#else
    float inv = 1.0f / fmaxf(sqrtf(nn), 1e-12f);
#endif
    r *= inv; x *= inv; y *= inv; z *= inv;

    float xx = x * x, yy = y * y, zz = z * z;
    float xy = x * y, xz = x * z, yz = y * z;
    float rx = r * x, ry = r * y, rz = r * z;

    float R00 = 1.0f - 2.0f * (yy + zz);
    float R01 = 2.0f * (xy - rz);
    float R02 = 2.0f * (xz + ry);
    float R10 = 2.0f * (xy + rz);
    float R11 = 1.0f - 2.0f * (xx + zz);
    float R12 = 2.0f * (yz - rx);
    float R20 = 2.0f * (xz - ry);
    float R21 = 2.0f * (yz + rx);
    float R22 = 1.0f - 2.0f * (xx + yy);

    float t0 = s0 * s0, t1 = s1 * s1, t2 = s2 * s2;

    float c00 = R00 * R00 * t0 + R01 * R01 * t1 + R02 * R02 * t2;
    float c01 = R00 * R10 * t0 + R01 * R11 * t1 + R02 * R12 * t2;
    float c02 = R00 * R20 * t0 + R01 * R21 * t1 + R02 * R22 * t2;
    float c11 = R10 * R10 * t0 + R11 * R11 * t1 + R12 * R12 * t2;
    float c12 = R10 * R20 * t0 + R11 * R21 * t1 + R12 * R22 * t2;
    float c22 = R20 * R20 * t0 + R21 * R21 * t1 + R22 * R22 * t2;

    c[0] = c00; c[1] = c01; c[2] = c02;
    c[3] = c01; c[4] = c11; c[5] = c12;
    c[6] = c02; c[7] = c12; c[8] = c22;
}

// Fast path: n is a multiple of TPB (true for N = 4,000,000).
// All global traffic is 128-bit and fully coalesced; scales + output are
// staged through LDS (bank-conflict-free strides 3 and 9, both coprime to 64),
// with the Global<->LDS staging done by the gfx1250 async DMA path (ASYNCcnt).
__global__ __launch_bounds__(TPB) void gauss_cov_fast(
    const float4* __restrict__ quat, const float* __restrict__ scales,
    float* __restrict__ out) {
    __shared__ alignas(16) float lds_s[TPB * 3];  // 3 KB staged scales
    __shared__ alignas(16) float lds_o[TPB * 9];  // 9 KB staged covariances

    const int tid  = threadIdx.x;
    const int base = blockIdx.x * TPB;

    // ---- Stage this block's scales (768 floats = 192 x b128) into LDS ----
    const float4* s4g = (const float4*)(scales + (size_t)base * 3);
    float4*       s4l = (float4*)lds_s;
#if USE_ASYNC_LDS
    for (int j = tid; j < (TPB * 3) / 4; j += TPB) {
        __builtin_amdgcn_global_load_async_to_lds_b128(
            (gv4i*)(s4g + j), (lv4i*)(s4l + j), 0, 0);
    }
#else
    for (int j = tid; j < (TPB * 3) / 4; j += TPB) s4l[j] = s4g[j];
#endif

    // Quaternion: naturally 16B/element -> direct coalesced b128 load.
    float4 q = quat[base + tid];

#if USE_ASYNC_LDS
    __builtin_amdgcn_s_wait_asynccnt(0);
#endif
    __syncthreads();

    float s0 = lds_s[tid * 3 + 0];
    float s1 = lds_s[tid * 3 + 1];
    float s2 = lds_s[tid * 3 + 2];

    float c[9];
    cov_from_quat(q.x, q.y, q.z, q.w, s0, s1, s2, c);

    // ---- Stage the 9 outputs in LDS, then drain with coalesced b128 ----
#pragma unroll
    for (int k = 0; k < 9; ++k) lds_o[tid * 9 + k] = c[k];
    __syncthreads();

    float4* o4g = (float4*)(out + (size_t)base * 9);
    float4* o4l = (float4*)lds_o;
#if USE_ASYNC_LDS
    for (int j = tid; j < (TPB * 9) / 4; j += TPB) {
        __builtin_amdgcn_global_store_async_from_lds_b128(
            (gv4i*)(o4g + j), (lv4i*)(o4l + j), 0, 0);
    }
    __builtin_amdgcn_s_wait_asynccnt(0);
#else
    for (int j = tid; j < (TPB * 9) / 4; j += TPB) o4g[j] = o4l[j];
#endif
}

// Generic path for arbitrary n (bounds-checked, scalar stores).
__global__ void gauss_cov_simple(const float4* __restrict__ quat,
                                 const float* __restrict__ scales,
                                 float* __restrict__ out, int n) {
    int i = blockIdx.x * blockDim.x + threadIdx.x;
    if (i >= n) return;
    float4 q = quat[i];
    float  c[9];
    cov_from_quat(q.x, q.y, q.z, q.w, scales[3 * i + 0], scales[3 * i + 1],
                  scales[3 * i + 2], c);
#pragma unroll
    for (int k = 0; k < 9; ++k) out[(size_t)i * 9 + k] = c[k];
}

extern "C" void kernel_launch(void* const* d_in, const int* in_sizes, int n_in,
                              void* d_out, int out_size, void* d_ws,
                              size_t ws_size, hipStream_t stream) {
    (void)n_in; (void)out_size; (void)d_ws; (void)ws_size;
    const float4* quat   = (const float4*)d_in[0];
    const float*  scales = (const float*)d_in[1];
    float*        out    = (float*)d_out;
    int n = in_sizes[0] / 4;  // quaternions are (N, 4) float32

    if (n > 0 && (n % TPB) == 0) {
        gauss_cov_fast<<<n / TPB, TPB, 0, stream>>>(quat, scales, out);
    } else if (n > 0) {
        gauss_cov_simple<<<(n + TPB - 1) / TPB, TPB, 0, stream>>>(quat, scales,
                                                                  out, n);
    }
}